// TDLoss_72567767433273
// MI455X (gfx1250) — compile-verified
//
#include <hip/hip_runtime.h>
#include <hip/hip_bf16.h>
#include <stdint.h>

// ---------------------------------------------------------------------------
// TD(lambda)/GAE loss, time-major [T=256, B=65536].
// Memory-bound (≈400 MB traffic, ≈0.17 GFLOP) -> design for 23.3 TB/s HBM:
//   * per-thread column, backward time scan in register
//   * CDNA5 async global->LDS (ASYNCcnt) 3-deep chunk pipeline for ≈20 MB of
//     bytes-in-flight device-wide (≈ BW x latency)
//   * non-temporal streaming stores
// ---------------------------------------------------------------------------

#define AS1 __attribute__((address_space(1)))
#define AS3 __attribute__((address_space(3)))

typedef int v4i __attribute__((vector_size(16)));

__device__ __forceinline__ void async_load_b128(const void* g, void* l) {
#if defined(__has_builtin) && __has_builtin(__builtin_amdgcn_global_load_async_to_lds_b128)
  // Builtin signature (probe-confirmed by diagnostic): (v4i AS1*, v4i AS3*, Ii, Ii)
  __builtin_amdgcn_global_load_async_to_lds_b128((AS1 v4i*)g, (AS3 v4i*)l, 0, 0);
#else
  // VDST = 32-bit LDS byte offset, VADDR = 64-bit global address.
  unsigned loff = (unsigned)(uintptr_t)l;  // low 32 bits of generic LDS ptr = LDS offset
  asm volatile("global_load_async_to_lds_b128 %0, %1, off" ::"v"(loff), "v"(g)
               : "memory");
#endif
}

template <int N>
__device__ __forceinline__ void wait_asynccnt() {
#if defined(__has_builtin) && __has_builtin(__builtin_amdgcn_s_wait_asynccnt)
  __builtin_amdgcn_s_wait_asynccnt(N);
#else
  asm volatile("s_wait_asynccnt %0" ::"i"(N) : "memory");
#endif
}

namespace {
constexpr int   T_STEPS = 256;              // time length (reference fixes T=256)
constexpr int   CH      = 8;                // rows per chunk
constexpr int   NCHUNKS = T_STEPS / CH;     // 32
constexpr int   COLS    = 256;              // columns per block == blockDim.x
constexpr int   NA      = 5;                // reward, step_type, discount, value, target_value
constexpr int   NBUF    = 3;                // pipeline depth (2 chunks in flight)
constexpr float kGamma  = 0.99f;
constexpr float kLambda = 0.95f;
}  // namespace

__global__ void __launch_bounds__(COLS)
gae_tdloss_kernel(const float* __restrict__ g_rw, const int* __restrict__ g_st,
                  const float* __restrict__ g_dc, const float* __restrict__ g_v,
                  const float* __restrict__ g_tv, float* __restrict__ g_out,
                  int n_cols) {
  __shared__ __align__(16) float smem[NBUF * NA * CH * COLS];  // 120 KB

  const int    tid   = threadIdx.x;
  const size_t B     = (size_t)n_cols;
  const int    col0  = blockIdx.x * COLS;
  const int    col   = col0 + tid;
  const int    rg    = tid >> 6;          // 0..3: which row of a 4-row group
  const int    lane4 = (tid & 63) << 2;   // element offset within a 256-wide row

  const char* gbase[NA] = {(const char*)g_rw, (const char*)g_st,
                           (const char*)g_dc, (const char*)g_v,
                           (const char*)g_tv};

  // Stage rows [c*CH, c*CH+8) of all 5 arrays for this block's 256 columns.
  // 256 threads x b128 = 4 rows per instruction -> 10 async ops per chunk.
  auto issue_chunk = [&](int c, int buf) {
    const int r0 = c * CH;
#pragma unroll
    for (int a = 0; a < NA; ++a) {
#pragma unroll
      for (int h = 0; h < 2; ++h) {
        const int   lr = h * 4 + rg;
        const char* g =
            gbase[a] + (((size_t)(r0 + lr) * B + (size_t)(col0 + lane4)) << 2);
        float* l = &smem[((buf * NA + a) * CH + lr) * COLS + lane4];
        async_load_b128(g, l);
      }
    }
  };

  // Row T-1 of the output is defined to be zero (tensor_extend_zero).
  __builtin_nontemporal_store(0.0f, &g_out[(size_t)(T_STEPS - 1) * B + col]);

  issue_chunk(NCHUNKS - 1, (NCHUNKS - 1) % NBUF);
  issue_chunk(NCHUNKS - 2, (NCHUNKS - 2) % NBUF);
  issue_chunk(NCHUNKS - 3, (NCHUNKS - 3) % NBUF);

  float adv = 0.0f;      // scan carry (adv_{t+1})
  float tv_next = 0.0f;  // target_value[t+1] carry
  float rw_b = 0.0f, dc_b = 0.0f;  // reward/discount of row r0+CH (chunk boundary)

  for (int c = NCHUNKS - 1; c >= 0; --c) {
    // Async loads complete in order per wave: <=20 outstanding means the two
    // youngest chunks may still be in flight, i.e. chunk c has landed.
    if (c >= 2)      wait_asynccnt<2 * NA * 2>();
    else if (c == 1) wait_asynccnt<NA * 2>();
    else             wait_asynccnt<0>();
    __syncthreads();  // all waves' async writes to this buffer are visible

    const int    buf = c % NBUF;
    const int    r0  = c * CH;
    const float* Lrw = &smem[((buf * NA + 0) * CH) * COLS];
    const int*   Lst = (const int*)&smem[((buf * NA + 1) * CH) * COLS];
    const float* Ldc = &smem[((buf * NA + 2) * CH) * COLS];
    const float* Lv  = &smem[((buf * NA + 3) * CH) * COLS];
    const float* Ltv = &smem[((buf * NA + 4) * CH) * COLS];

    // One backward step at time t = r0 + lr, needing row t+1's reward/discount.
    auto do_step = [&](int lr, float rw_n, float dc_n) {
      const float tv_t  = Ltv[lr * COLS + tid];
      const float v_t   = Lv [lr * COLS + tid];
      const int   st_t  = Lst[lr * COLS + tid];
      const float dg    = dc_n * kGamma;                   // disc_{t+1}
      const float delta = fmaf(dg, tv_next, rw_n) - tv_t;  // r + disc*V' - V
      const float advn  = fmaf(dg * kLambda, adv, delta);  // delta + wdisc*adv'
      adv = (st_t == 2) ? 0.0f : advn;                     // (1 - is_last)*(...)
      const float diff = v_t - (adv + tv_t);               // value - returns
      __builtin_nontemporal_store(diff * diff,
                                  &g_out[(size_t)(r0 + lr) * B + col]);
      tv_next = tv_t;
    };

    if (c == NCHUNKS - 1) {
      // Topmost chunk: no step at t = T-1; seed tv carry with V_{T-1}.
      tv_next = Ltv[(CH - 1) * COLS + tid];
    } else {
      do_step(CH - 1, rw_b, dc_b);  // t = r0+7 uses row r0+8 boundary registers
    }
#pragma unroll
    for (int lr = CH - 2; lr >= 0; --lr)
      do_step(lr, Lrw[(lr + 1) * COLS + tid], Ldc[(lr + 1) * COLS + tid]);

    rw_b = Lrw[tid];  // row r0 -> boundary for chunk c-1
    dc_b = Ldc[tid];
    __syncthreads();  // everyone is done reading this buffer

    if (c >= NBUF) issue_chunk(c - NBUF, buf);  // refill the buffer just freed
  }
}

extern "C" void kernel_launch(void* const* d_in, const int* in_sizes, int n_in,
                              void* d_out, int out_size, void* d_ws,
                              size_t ws_size, hipStream_t stream) {
  (void)n_in; (void)out_size; (void)d_ws; (void)ws_size;
  const float* rw = (const float*)d_in[0];
  const int*   st = (const int*)d_in[1];
  const float* dc = (const float*)d_in[2];
  const float* v  = (const float*)d_in[3];
  const float* tv = (const float*)d_in[4];
  float*       out = (float*)d_out;

  const int B      = in_sizes[0] / T_STEPS;  // time-major [T, B], T = 256
  const int blocks = B / COLS;               // 65536/256 = 256 blocks x 8 waves

  hipLaunchKernelGGL(gae_tdloss_kernel, dim3(blocks), dim3(COLS), 0, stream,
                     rw, st, dc, v, tv, out, B);
}